// QuantLSTMCell_2224793060103
// MI455X (gfx1250) — compile-verified
//
#include <hip/hip_runtime.h>
#include <math.h>

typedef float v2f __attribute__((ext_vector_type(2)));
typedef float v8f __attribute__((ext_vector_type(8)));

#define DDIM 4096
#define KDIM 8192
#define KCH  2048      // K elements per split-K chunk
#define NSPLIT 4
#define QMAXF 127.0f

// ---------- helpers ----------

__device__ __forceinline__ float dq(float v, float m) {
  // 8-bit symmetric fake quant, matches bits_quant():
  // scale = m/127 if m>0 else 1 ; clip(round(v/scale), -127, 127)*scale
  float s = (m > 0.0f) ? (m / QMAXF) : 1.0f;
  float r = rintf(v / s);                       // round-half-even, like jnp.round
  r = fminf(fmaxf(r, -QMAXF), QMAXF);
  return r * s;
}

// wave32 max-abs reduction, then ONE atomic per wave (avoids same-address
// atomic serialization in L2; atomicMax on the uint bit pattern is exact and
// order-independent -> deterministic).
__device__ __forceinline__ void amax_wave(float* slot, float v) {
  float m = fabsf(v);
#pragma unroll
  for (int off = 16; off > 0; off >>= 1)
    m = fmaxf(m, __shfl_xor(m, off, 32));
  if ((threadIdx.x & 31) == 0)
    atomicMax(reinterpret_cast<unsigned int*>(slot), __float_as_uint(m));
}

__device__ __forceinline__ float sigm(float x) {
  return 1.0f / (1.0f + expf(-x));
}

// ---------- small kernels ----------

__global__ void k_init(float* scal) {
  if (threadIdx.x < 32) scal[threadIdx.x] = 0.0f;
}

__global__ void k_absmax(const float* __restrict__ a, int n, float* slot) {
  int i = blockIdx.x * blockDim.x + threadIdx.x;
  float v = (i < n) ? a[i] : 0.0f;
  amax_wave(slot, v);
}

// abs-max of each gate's bias -> scal[5+g]   (grid.y = gate)
__global__ void k_absmax_bias(const float* b0, const float* b1,
                              const float* b2, const float* b3, float* scal) {
  int g = blockIdx.y;
  int i = blockIdx.x * blockDim.x + threadIdx.x;
  const float* b = (g == 0) ? b0 : (g == 1) ? b1 : (g == 2) ? b2 : b3;
  amax_wave(&scal[5 + g], b[i]);
}

// concat = Q([h, x]) with one shared scale (scal[0])
__global__ void k_quant_concat(const float* __restrict__ h,
                               const float* __restrict__ x,
                               const float* __restrict__ scal,
                               float* __restrict__ cq) {
  int i = blockIdx.x * blockDim.x + threadIdx.x;
  if (i < KDIM) {
    float v = (i < DDIM) ? h[i] : x[i - DDIM];
    cq[i] = dq(v, scal[0]);
  }
}

// ---------- the big one: 4-gate GEMV via V_WMMA_F32_16X16X4_F32 ----------
// grid = (DDIM/128, 4 gates, NSPLIT k-chunks), block = 256 (8 waves)
// wave w of block bx handles columns [ (bx*8+w)*16 , +16 )
__global__ void k_gemv_wmma(const float* __restrict__ cq,
                            const float* __restrict__ w0,  // gate 0: Wc
                            const float* __restrict__ w1,  // gate 1: Wf
                            const float* __restrict__ w2,  // gate 2: Wi
                            const float* __restrict__ w3,  // gate 3: Wo
                            float* __restrict__ part) {
  __shared__ float lx[KCH];

  const int gate = blockIdx.y;
  const float* W = (gate == 0) ? w0 : (gate == 1) ? w1 : (gate == 2) ? w2 : w3;
  const int kc     = blockIdx.z;
  const int k_base = kc * KCH;

  // stage the quantized activation chunk in LDS (8 KB, one pass, uniform loop)
  for (int i = threadIdx.x; i < KCH; i += blockDim.x) lx[i] = cq[k_base + i];
  __syncthreads();

  const int wave = threadIdx.x >> 5;
  const int lane = threadIdx.x & 31;
  const int n0   = (blockIdx.x * 8 + wave) * 16;
  const int col  = n0 + (lane & 15);
  // ISA A/B 16x4 fp32 layout: lanes 0-15 carry K={0,1}, lanes 16-31 carry K={2,3}
  const int kofs = (lane < 16) ? 0 : 2;

  const float* wp = W + (size_t)(k_base + kofs) * DDIM + col;

  v8f acc = {};
#pragma unroll 8
  for (int kk = 0; kk < KCH; kk += 4) {
    v2f a, b;
    a.x = lx[kk + kofs];         // A[m][K=kofs]   (same value for every m-row)
    a.y = lx[kk + kofs + 1];     // A[m][K=kofs+1]
    b.x = wp[0];                 // B[K=kofs  ][n] = W[k][col]      (64B coalesced)
    b.y = wp[DDIM];              // B[K=kofs+1][n] = W[k+1][col]
    wp += (size_t)4 * DDIM;
    acc = __builtin_amdgcn_wmma_f32_16x16x4_f32(
        /*neg_a=*/false, a, /*neg_b=*/false, b,
        /*c_mod=*/(short)0, acc, /*reuse_a=*/false, /*reuse_b=*/false);
  }

  // D row 0 lives in acc[0] on lanes 0-15 (M=0, N=lane); all rows identical.
  if (lane < 16)
    part[((size_t)gate * NSPLIT + kc) * DDIM + col] = acc[0];
}

// raw[g][n] = sum of split-K partials; fold abs-max into scal[1+g]
__global__ void k_reduce_part(const float* __restrict__ part,
                              float* __restrict__ raw, float* scal) {
  int g = blockIdx.y;
  int n = blockIdx.x * blockDim.x + threadIdx.x;
  const float* p = part + (size_t)g * NSPLIT * DDIM;
  float s = p[n] + p[DDIM + n] + p[2 * DDIM + n] + p[3 * DDIM + n];
  raw[(size_t)g * DDIM + n] = s;
  amax_wave(&scal[1 + g], s);
}

// y = Q(raw, rawmax) + Q(b, bmax); track |y| max in scal[9+g]
__global__ void k_combine(float* __restrict__ raw,
                          const float* b0, const float* b1,
                          const float* b2, const float* b3, float* scal) {
  int g = blockIdx.y;
  int n = blockIdx.x * blockDim.x + threadIdx.x;
  const float* b = (g == 0) ? b0 : (g == 1) ? b1 : (g == 2) ? b2 : b3;
  size_t idx = (size_t)g * DDIM + n;
  float y = dq(raw[idx], scal[1 + g]) + dq(b[n], scal[5 + g]);
  raw[idx] = y;
  amax_wave(&scal[9 + g], y);
}

// a = act(Q(y)); track |a| max in scal[13+g]   (gate 0 = tanh, others sigmoid)
__global__ void k_act(const float* __restrict__ raw,
                      float* __restrict__ act, float* scal) {
  int g = blockIdx.y;
  int n = blockIdx.x * blockDim.x + threadIdx.x;
  float q = dq(raw[(size_t)g * DDIM + n], scal[9 + g]);
  float a = (g == 0) ? tanhf(q) : sigm(q);
  act[(size_t)g * DDIM + n] = a;
  amax_wave(&scal[13 + g], a);
}

__global__ void k_qact(float* __restrict__ act, const float* __restrict__ scal) {
  int g = blockIdx.y;
  int n = blockIdx.x * blockDim.x + threadIdx.x;
  size_t idx = (size_t)g * DDIM + n;
  act[idx] = dq(act[idx], scal[13 + g]);
}

// tail: memory = Q(z*zi); c_new = Q(Q(c*zf)+memory); h = Q(zo*Q(tanh(c_new)))
__global__ void k_t1(const float* __restrict__ act, const float* __restrict__ c,
                     float* p1, float* p2, float* scal) {
  int i = blockIdx.x * blockDim.x + threadIdx.x;
  float z = act[i], zf = act[DDIM + i], zi = act[2 * DDIM + i];
  float a = z * zi;       // -> memory (pre-quant)
  float b = c[i] * zf;    // -> c*zf   (pre-quant)
  p1[i] = a; p2[i] = b;
  amax_wave(&scal[17], a);
  amax_wave(&scal[18], b);
}

__global__ void k_t2(const float* p1, const float* p2, float* s, float* scal) {
  int i = blockIdx.x * blockDim.x + threadIdx.x;
  float v = dq(p2[i], scal[18]) + dq(p1[i], scal[17]);
  s[i] = v;
  amax_wave(&scal[19], v);
}

__global__ void k_t3(const float* s, float* t, float* scal) {
  int i = blockIdx.x * blockDim.x + threadIdx.x;
  float v = tanhf(dq(s[i], scal[19]));
  t[i] = v;
  amax_wave(&scal[20], v);
}

__global__ void k_t4(const float* __restrict__ act, const float* t,
                     float* hp, float* scal) {
  int i = blockIdx.x * blockDim.x + threadIdx.x;
  float v = act[3 * DDIM + i] * dq(t[i], scal[20]);
  hp[i] = v;
  amax_wave(&scal[21], v);
}

__global__ void k_t5(const float* hp, const float* scal, float* out) {
  int i = blockIdx.x * blockDim.x + threadIdx.x;
  out[i] = dq(hp[i], scal[21]);
}

// ---------- launch ----------

extern "C" void kernel_launch(void* const* d_in, const int* in_sizes, int n_in,
                              void* d_out, int out_size, void* d_ws, size_t ws_size,
                              hipStream_t stream) {
  (void)in_sizes; (void)n_in; (void)out_size; (void)ws_size;

  const float* x  = (const float*)d_in[0];
  const float* c  = (const float*)d_in[1];
  const float* h  = (const float*)d_in[2];
  const float* Wf = (const float*)d_in[3];
  const float* bf = (const float*)d_in[4];
  const float* Wi = (const float*)d_in[5];
  const float* bi = (const float*)d_in[6];
  const float* Wc = (const float*)d_in[7];
  const float* bc = (const float*)d_in[8];
  const float* Wo = (const float*)d_in[9];
  const float* bo = (const float*)d_in[10];
  float* out = (float*)d_out;

  // workspace layout (floats)
  float* ws   = (float*)d_ws;
  float* CQ   = ws;                    //  8192  quantized concat [h, x]
  float* PART = ws + 8192;             // 65536  split-K partials (reused below)
  float* RAW  = ws + 73728;            // 16384  raw / pre-act per gate
  float* ACT  = ws + 90112;            // 16384  activated gates z,zf,zi,zo
  float* SCAL = ws + 106496;           //    32  reduction scalars
  float* P1   = PART;                  // z*zi       (PART free after reduce)
  float* P2   = PART + 4096;           // c*zf
  float* S    = PART + 8192;           // Q(c*zf)+memory
  float* T    = PART + 12288;          // tanh(c_new)
  float* HP   = PART + 16384;          // zo * Q(tanh(c_new))

  const dim3 B(256);
  const dim3 G16(16);                  // 4096 elements
  const dim3 G16x4(16, 4);             // 4 gates x 4096

  k_init<<<1, 32, 0, stream>>>(SCAL);

  // concat scale = max|[h, x]|
  k_absmax<<<G16, B, 0, stream>>>(h, DDIM, &SCAL[0]);
  k_absmax<<<G16, B, 0, stream>>>(x, DDIM, &SCAL[0]);
  k_quant_concat<<<dim3(32), B, 0, stream>>>(h, x, SCAL, CQ);

  // 4 GEMVs (gate order: 0=c/tanh, 1=f, 2=i, 3=o), split-K = 4
  k_gemv_wmma<<<dim3(DDIM / 128, 4, NSPLIT), B, 0, stream>>>(
      CQ, Wc, Wf, Wi, Wo, PART);
  k_reduce_part<<<G16x4, B, 0, stream>>>(PART, RAW, SCAL);

  // bias scales (one kernel, gate in grid.y)
  k_absmax_bias<<<G16x4, B, 0, stream>>>(bc, bf, bi, bo, SCAL);

  k_combine<<<G16x4, B, 0, stream>>>(RAW, bc, bf, bi, bo, SCAL);
  k_act<<<G16x4, B, 0, stream>>>(RAW, ACT, SCAL);
  k_qact<<<G16x4, B, 0, stream>>>(ACT, SCAL);

  // cell-state tail
  k_t1<<<G16, B, 0, stream>>>(ACT, c, P1, P2, SCAL);
  k_t2<<<G16, B, 0, stream>>>(P1, P2, S, SCAL);
  k_t3<<<G16, B, 0, stream>>>(S, T, SCAL);
  k_t4<<<G16, B, 0, stream>>>(ACT, T, HP, SCAL);
  k_t5<<<G16, B, 0, stream>>>(HP, SCAL, out);
}